// Zernike_87402584473694
// MI455X (gfx1250) — compile-verified
//
#include <hip/hip_runtime.h>
#include <hip/hip_bf16.h>
#include <math.h>

// ---------------------------------------------------------------------------
// Zernike 5x5 conv, NCHW, implicit GEMM on CDNA5 WMMA (f32 16x16x4).
//   x   : [B=8, IC=8, H=1024, W=1024] f32
//   out : [B=8, OC=16, H=1024, W=1024] f32
//   GEMM: D[16 oc, 16 px] += A[16 oc, K] * B[K, 16 px], K = 200 (50 steps x 4)
//
// K-slot permutation trick: WMMA's K-reduction is order-invariant, so each
// step's 4 hardware K-slots (h = lane>>4, v = vgpr) are assigned logical
// k = (ic,fy,fx) such that the h=0 / h=1 taps differ by a constant LDS
// delta (1, TW, or TH*TW elements).  "half" then folds into one of three
// per-lane base pointers and every B load is ds_load_b32 with an immediate
// offset -- zero in-loop address VALU, no indirection table.
// ---------------------------------------------------------------------------

#define BATCH   8
#define IC      8
#define OC      16
#define IMH     1024
#define IMW     1024
#define FILT    5
#define KTOT    (IC * FILT * FILT)   // 200
#define KSTEPS  (KTOT / 4)           // 50 WMMA steps of K=4

// conv-kernel tiling: block = 16 waves = 512 threads -> 4 rows x 4 col-tiles
#define ROWS    4
#define COLS    64
#define TH      (ROWS + 4)           // 8 input rows staged
#define TCW     (COLS + 4)           // 68 valid input cols
#define TW      72                   // padded LDS row stride (floats)
#define BLOCK   512

typedef __attribute__((ext_vector_type(2))) float v2f;
typedef __attribute__((ext_vector_type(8))) float v8f;

// step s, half h, vgpr v  ->  logical k = ic*25 + fy*5 + fx
//   s <  40 : fx-quad  (ic = s/5, fy = s%5, fx = 2v+h)
//   s <  48 : fy-quad  (ic = s-40, fy = 2v+h, fx = 4)
//   else    : ic-quad  (ic = (s-48)*4 + 2v+h, fy = 4, fx = 4)
__host__ __device__ constexpr int kmap(int s, int h, int v) {
    return (s < 40) ? ((s / 5) * 25 + (s % 5) * 5 + 2 * v + h)
         : (s < 48) ? ((s - 40) * 25 + (2 * v + h) * 5 + 4)
                    : (((s - 48) * 4 + 2 * v + h) * 25 + 24);
}

// B-fragment LDS element offset for (s, v), EXCLUDING the h contribution
// (h is folded into the per-lane base pointer: +1, +TW, or +TH*TW)
__host__ __device__ constexpr int boff(int s, int v) {
    return (s < 40) ? ((s / 5) * (TH * TW) + (s % 5) * TW + 2 * v)
         : (s < 48) ? ((s - 40) * (TH * TW) + (2 * v) * TW + 4)
                    : (((s - 48) * 4 + 2 * v) * (TH * TW) + 4 * TW + 4);
}

// ---------------------------------------------------------------------------
// Kernel 1: evaluate Zernike weights, emit A-fragments pre-swizzled with the
// SAME kmap:  Apack[(s*32 + lane)*2 + v] = W[m = lane&15][kmap(s, lane>>4, v)]
// ---------------------------------------------------------------------------
__global__ __launch_bounds__(128) void zernike_weights_kernel(
    const float* __restrict__ coeff,   // [128, S]
    const int*   __restrict__ expn,    // [128, S]
    const int*   __restrict__ nvals,   // [128]
    int S,
    float* __restrict__ apack)         // [KSTEPS*32*2] floats
{
    __shared__ float Wmat[OC * KTOT];  // [16][200]

    const int p = threadIdx.x;         // pair 0..127
    const float CENTER = 2.0f;         // FILT/2 - 0.5
    const float CUTOFF = sqrtf(CENTER * CENTER + 0.25f);
    const float LNE    = logf(2.71828f);

    if (p < 128) {
        const int oc = p >> 3;
        const int ic = p & 7;
        const float nf = (float)nvals[p];
        for (int i = 0; i < FILT; ++i) {
            for (int j = 0; j < FILT; ++j) {
                float a = fabsf((float)i - CENTER);
                float b = fabsf((float)j - CENTER);
                float dist = sqrtf(a * a + b * b);
                float radial = (dist <= CUTOFF) ? dist : 0.0f;
                float rsafe  = (radial > 0.0f) ? radial : 1.0f;
                float s = 0.0f;
                for (int q = 0; q < S; ++q) {
                    float c = coeff[p * S + q];
                    float e = (float)expn[p * S + q];
                    s += c * powf(rsafe, e);
                }
                float poly = (radial > 0.0f) ? s : 0.0f;
                float phi  = atanf(b / (a + 0.001f));
                float cosr = cosf(nf * phi * LNE);
                float w = (poly > 0.0f) ? poly * cosr : poly;
                Wmat[oc * KTOT + ic * 25 + i * 5 + j] = w;
            }
        }
    }
    __syncthreads();

    // swizzle into WMMA A-fragment order using kmap
    for (int i = threadIdx.x; i < KSTEPS * 32 * 2; i += 128) {
        int s = i >> 6;            // step
        int l = (i >> 1) & 31;     // lane
        int v = i & 1;             // vgpr within pair
        int h = l >> 4;
        int m = l & 15;
        apack[i] = Wmat[m * KTOT + kmap(s, h, v)];
    }
}

// ---------------------------------------------------------------------------
// Kernel 2: implicit-GEMM conv. One wave -> 16oc x 16px tile, 50 WMMAs.
// ---------------------------------------------------------------------------
__global__ __launch_bounds__(BLOCK) void zernike_conv_kernel(
    const float* __restrict__ x,
    const float* __restrict__ apack,
    float* __restrict__ out)
{
    __shared__ float  xs[IC * TH * TW];     // input tile w/ zero halo (18.4 KB)
    __shared__ float2 As[KSTEPS * 32];      // A fragments            (12.8 KB)

    const int tid = threadIdx.x;

    // block -> (batch, row-tile, col-tile)
    const int tilesX = IMW / COLS;          // 16
    const int tilesY = IMH / ROWS;          // 256
    int bid = blockIdx.x;
    const int cx = bid % tilesX;  bid /= tilesX;
    const int cy = bid % tilesY;  bid /= tilesY;
    const int b  = bid;
    const int x0 = cx * COLS;
    const int y0 = cy * ROWS;

    // ---- stage input tile (zero-filled halo), coalesced along columns ----
    for (int i = tid; i < IC * TH * TCW; i += BLOCK) {
        int c  = i % TCW;
        int t  = i / TCW;
        int r  = t % TH;
        int ic = t / TH;
        int gy = y0 + r - 2;
        int gx = x0 + c - 2;
        float v = 0.0f;
        if ((unsigned)gy < IMH && (unsigned)gx < IMW)
            v = x[(((size_t)b * IC + ic) * IMH + gy) * IMW + gx];
        xs[ic * (TH * TW) + r * TW + c] = v;
    }

    // ---- stage pre-swizzled A fragments (contiguous copy) ----
    {
        const float2* ap = (const float2*)apack;
        for (int i = tid; i < KSTEPS * 32; i += BLOCK) As[i] = ap[i];
    }

    __syncthreads();

    // ---- per-wave 16x16 tile ----
    const int wave = tid >> 5;              // 0..15 (wave32)
    const int lane = tid & 31;
    const int half = lane >> 4;
    const int n    = lane & 15;             // pixel column within tile
    const int wrow = wave >> 2;             // 0..3 output row in block
    const int wcol = (wave & 3) * 16;       // column sub-tile

    // per-lane base (row/col of this lane's B column) + half-delta per class
    const int bbase = wrow * TW + wcol + n;
    const float* pFx = xs + bbase + half;               // fx-quads: delta 1
    const float* pFy = xs + bbase + half * TW;          // fy-quads: delta TW
    const float* pIc = xs + bbase + half * (TH * TW);   // ic-quads: delta TH*TW
    const float2* Alane = As + lane;

    v8f acc = {0.f, 0.f, 0.f, 0.f, 0.f, 0.f, 0.f, 0.f};

#pragma unroll
    for (int s = 0; s < KSTEPS; ++s) {
        const float* pb = (s < 40) ? pFx : ((s < 48) ? pFy : pIc);
        v2f bfrag;
        bfrag.x = pb[boff(s, 0)];                   // ds_load_b32 offset:imm
        bfrag.y = pb[boff(s, 1)];                   // ds_load_b32 offset:imm
        float2 af = Alane[s * 32];                  // ds_load_b64 offset:imm
        v2f afrag;
        afrag.x = af.x;
        afrag.y = af.y;
        acc = __builtin_amdgcn_wmma_f32_16x16x4_f32(
            /*neg_a=*/false, afrag, /*neg_b=*/false, bfrag,
            /*c_mod=*/(short)0, acc, /*reuse_a=*/false, /*reuse_b=*/false);
    }

    // ---- store: C/D layout -> lane l holds col n, VGPR r holds oc = r + 8*half
    const int oy = y0 + wrow;
    float* outp = out + (size_t)b * OC * IMH * IMW
                      + (size_t)oy * IMW + x0 + wcol + n;
#pragma unroll
    for (int r = 0; r < 8; ++r) {
        int oc = r + 8 * half;
        outp[(size_t)oc * (IMH * IMW)] = acc[r];
    }
}

// ---------------------------------------------------------------------------
extern "C" void kernel_launch(void* const* d_in, const int* in_sizes, int n_in,
                              void* d_out, int out_size, void* d_ws, size_t ws_size,
                              hipStream_t stream) {
    const float* x     = (const float*)d_in[0];
    const float* coeff = (const float*)d_in[1];
    const int*   expn  = (const int*)d_in[2];
    const int*   nvals = (const int*)d_in[3];
    float*       out   = (float*)d_out;
    float*       apack = (float*)d_ws;            // KSTEPS*32*2 floats = 12.8 KB

    const int S = in_sizes[1] / (OC * IC);        // coeff is [128, S]

    zernike_weights_kernel<<<1, 128, 0, stream>>>(coeff, expn, nvals, S, apack);

    const int nBlocks = BATCH * (IMH / ROWS) * (IMW / COLS);   // 32768
    zernike_conv_kernel<<<nBlocks, BLOCK, 0, stream>>>(x, apack, out);
}